// RBFFeatExtractLayer_38989713113611
// MI455X (gfx1250) — compile-verified
//
#include <hip/hip_runtime.h>

// RBF feature extraction: out[b,m,k] = exp(-||X[b,m,:] - Q[k,:]||^2 / omega[k]^2)
// B=16, M=32768, NX=3, K=85.  Output 178 MB f32 -> store-bandwidth bound
// (~7.7 us at 23.3 TB/s). The pairwise-distance cross term is routed through
// V_WMMA_F32_16X16X4_F32 via the identity
//   ||x-q||^2 = (xx - 2 x.q) + qq,  A-row = [x0,x1,x2,xx], B-col = [-2q0,-2q1,-2q2,1]
// so one WMMA yields (xx - 2 x.q) for a 16x16 tile; qq and -1/w^2 are
// per-column scalars, which in the C/D layout are per-lane constants.

typedef float v2f __attribute__((ext_vector_type(2)));
typedef float v8f __attribute__((ext_vector_type(8)));

#define BATCH   16
#define M_DIM   32768
#define K_DIM   85
#define NROWS   (BATCH * M_DIM)     // 524288 rows of X / out
#define N_CTILE 6                   // ceil(85/16) column tiles (padded to 96)
#define WAVES_PER_BLOCK 8

__global__ __launch_bounds__(256)
void rbf_wmma_kernel(const float* __restrict__ X,      // [NROWS, 3]
                     const float* __restrict__ Q,      // [K_DIM, 3]
                     const float* __restrict__ omega,  // [K_DIM]
                     float* __restrict__ out)          // [NROWS, K_DIM]
{
    const int lane = threadIdx.x & 31;
    const int wave = threadIdx.x >> 5;
    const int tile = blockIdx.x * WAVES_PER_BLOCK + wave;  // 16-row tile
    const int r0   = tile * 16;

    // ---- A fragment: 16x4 f32. lane holds row M = lane%16.
    // lanes 0-15: VGPR0=K0, VGPR1=K1 ; lanes 16-31: VGPR0=K2, VGPR1=K3.
    // K-slots: [x0, x1, x2, xx] with xx = ||x||^2.
    const int   hi   = lane >> 4;            // 0 -> K pair {0,1}, 1 -> {2,3}
    const int   arow = r0 + (lane & 15);
    const float x0 = X[arow * 3 + 0];
    const float x1 = X[arow * 3 + 1];
    const float x2 = X[arow * 3 + 2];
    const float xx = x0 * x0 + x1 * x1 + x2 * x2;
    v2f a;
    a.x = hi ? x2 : x0;
    a.y = hi ? xx : x1;

    // ---- loop over 6 column tiles of 16 queries each ----
    for (int ct = 0; ct < N_CTILE; ++ct) {
        const int   col   = ct * 16 + (lane & 15);
        const int   cc    = (col < K_DIM) ? col : (K_DIM - 1);  // clamp OOB reads
        const float valid = (col < K_DIM) ? 1.0f : 0.0f;

        const float q0 = Q[cc * 3 + 0];
        const float q1 = Q[cc * 3 + 1];
        const float q2 = Q[cc * 3 + 2];
        const float w  = omega[cc];

        // B fragment: 4x16 f32. lane holds col N = lane%16.
        // lanes 0-15: VGPR0=K0, VGPR1=K1 ; lanes 16-31: VGPR0=K2, VGPR1=K3.
        // K-slots: [-2q0, -2q1, -2q2, 1]; padded columns zeroed.
        v2f b;
        b.x = (hi ? (-2.0f * q2) : (-2.0f * q0)) * valid;
        b.y = (hi ? 1.0f         : (-2.0f * q1)) * valid;

        v8f c = {};
        // 8 args: (neg_a, A, neg_b, B, c_mod, C, reuse_a, reuse_b)
        c = __builtin_amdgcn_wmma_f32_16x16x4_f32(
                /*neg_a=*/false, a, /*neg_b=*/false, b,
                /*c_mod=*/(short)0, c, /*reuse_a=*/false, /*reuse_b=*/false);
        // c[v] = xx - 2 x.q  for row m = r0 + v + hi*8, col = ct*16 + lane%16

        if (col < K_DIM) {
            const float qq     = q0 * q0 + q1 * q1 + q2 * q2;
            // single v_rcp_f32 instead of the IEEE divide expansion
            const float ninvw2 = -__builtin_amdgcn_rcpf(w * w);
#pragma unroll
            for (int v = 0; v < 8; ++v) {
                const int m = r0 + v + hi * 8;
                const float d2 = c[v] + qq;               // ||x-q||^2
                out[(size_t)m * K_DIM + col] = __expf(d2 * ninvw2);
            }
        }
    }
}

extern "C" void kernel_launch(void* const* d_in, const int* in_sizes, int n_in,
                              void* d_out, int out_size, void* d_ws, size_t ws_size,
                              hipStream_t stream)
{
    const float* X     = (const float*)d_in[0];   // [16, 32768, 3]
    const float* Q     = (const float*)d_in[1];   // [85, 3]
    const float* omega = (const float*)d_in[2];   // [85]
    float* out         = (float*)d_out;           // [16, 32768, 85]

    const int n_tiles = NROWS / 16;                       // 32768 row tiles
    const int blocks  = n_tiles / WAVES_PER_BLOCK;        // 4096 blocks
    rbf_wmma_kernel<<<blocks, 256, 0, stream>>>(X, Q, omega, out);
}